// ConvolutionVoxel_82514911691298
// MI455X (gfx1250) — compile-verified
//
#include <hip/hip_runtime.h>

typedef __attribute__((ext_vector_type(2))) float v2f;
typedef __attribute__((ext_vector_type(8))) float v8f;

#define CCH 32                    // channels = MUL + 3*MUL
#define XT 32                     // x-tile width per block
#define XW 36                     // staged row width (XT + 4 halo)
#define PLANEF (16 * 5 * XW * 2)  // floats per staged plane: [cpair][yy][xx][2] = 5760
// dynamic LDS: 2 * PLANEF * 4 = 46080 bytes

__device__ __forceinline__ float soft_unit_step(float t) {
    return t > 0.f ? __expf(-1.f / t) : 0.f;
}

__device__ __forceinline__ void async_copy_b32(const float* __restrict__ g, float* l) {
#if __has_builtin(__builtin_amdgcn_global_load_async_to_lds_b32)
    __builtin_amdgcn_global_load_async_to_lds_b32(
        (__attribute__((address_space(1))) int*)g,
        (__attribute__((address_space(3))) int*)l, 0, 0);
#else
    unsigned loff = (unsigned)(size_t)(__attribute__((address_space(3))) float*)l;
    asm volatile("global_load_async_to_lds_b32 %0, %1, off"
                 :: "v"(loff), "v"((unsigned long long)(size_t)g)
                 : "memory");
#endif
}

__device__ __forceinline__ void wait_async0() {
#if __has_builtin(__builtin_amdgcn_s_wait_asynccnt)
    __builtin_amdgcn_s_wait_asynccnt(0);
#else
    asm volatile("s_wait_asynccnt 0" ::: "memory");
#endif
}

// ---------------------------------------------------------------------------
// Build the fused conv kernel (incl. self-connection folded into center tap)
// directly in WMMA A-fragment layout:
//   Aw2[(((t*8 + c)*2 + mt)*32 + lane)] = float2{ A(m,kk), A(m,kk+1) }
//   m = lane%16 (+16*mt -> output channel), kk = 2*(lane/16),
//   input channel i = 4*c + kk + j, tap t = dz*25 + dy*5 + dx.
// ---------------------------------------------------------------------------
__global__ void build_weights_kernel(const float* __restrict__ tp,
                                     const float* __restrict__ wsc0,
                                     const float* __restrict__ wsc1,
                                     float* __restrict__ Aw) {
    __shared__ float wsh[256];
    const int t  = blockIdx.x;            // 0..124
    const int dz = t / 25, dy = (t / 5) % 5, dx = t % 5;
    const float ra = -1.f + 0.5f * (float)dz;
    const float rb = -1.f + 0.5f * (float)dy;
    const float rc = -1.f + 0.5f * (float)dx;
    const float n  = sqrtf(ra * ra + rb * rb + rc * rc);

    float emb[5];
    #pragma unroll
    for (int m = 0; m < 5; ++m) {
        float diff = 6.f * n - (float)(m + 1);
        emb[m] = 8.4335731f * soft_unit_step(diff + 1.f) * soft_unit_step(1.f - diff);
    }
    const float invn = 1.f / fmaxf(n, 1e-12f);
    float sh1[3];                          // sqrt(3) * (unit_y, unit_z, unit_x)
    sh1[0] = 1.7320508f * rb * invn;
    sh1[1] = 1.7320508f * rc * invn;
    sh1[2] = 1.7320508f * ra * invn;

    const int u = threadIdx.x;             // 256 threads -> one column each
    float acc = 0.f;
    #pragma unroll
    for (int m = 0; m < 5; ++m) acc += emb[m] * tp[m * 256 + u];
    wsh[u] = acc * 0.0894427191f;          // / 5^1.5
    __syncthreads();

    for (int q = 0; q < 4; ++q) {
        int idx = threadIdx.x + q * 256;
        int o = idx >> 5;                  // output channel (column of K)
        int i = idx & 31;                  // input channel  (row of K)
        float val;
        if (o < 8) {
            if (i < 8) {
                val = 0.25f * wsh[0 * 64 + i * 8 + o];                      // B00 (W1)
                if (t == 62) val += 3.5355339f * wsc0[i * 8 + o];           // 10/sqrt(8)*sc0
            } else {
                int uu = (i - 8) / 3, ii = (i - 8) % 3;
                val = 0.14433757f * wsh[3 * 64 + uu * 8 + o] * sh1[ii];     // B10 (W4)
            }
        } else {
            int w_ = (o - 8) / 3, k = (o - 8) % 3;
            if (i < 8) {
                val = 0.25f * wsh[1 * 64 + i * 8 + w_] * sh1[k];            // B01 (W2)
            } else {
                int uu = (i - 8) / 3, ii = (i - 8) % 3;
                val = (ii == k) ? 0.25f * wsh[2 * 64 + uu * 8 + w_] : 0.f;  // B11 (W3)
                if (t == 62 && ii == k) val += 3.5355339f * wsc1[uu * 8 + w_];
            }
        }
        int c = i >> 2, rr = i & 3;
        int hi = rr >> 1, j = rr & 1;
        int lane = hi * 16 + (o & 15);
        int mt = o >> 4;
        Aw[((((t * 8 + c) * 2 + mt) * 32 + lane) << 1) + j] = val;
    }
}

// ---------------------------------------------------------------------------
// Implicit-GEMM 5x5x5 conv via V_WMMA_F32_16X16X4_F32.
// Block: 128 threads = 4 waves = 2 M-tiles (o) x 2 N-tiles (x).
// Grid: (y, z, b*2 + x-half). Double-buffered halo planes staged with
// GLOBAL_LOAD_ASYNC_TO_LDS_B32 (ASYNCcnt) overlapped with the WMMA schedule.
// LDS plane layout: [cpair 0..15][yy 0..4][xx 0..35][2] -> one ds_load_b64
// fetches both K-rows of a B fragment per lane.
// ---------------------------------------------------------------------------
__global__ __launch_bounds__(128)
void conv_wmma_kernel(const float* __restrict__ x,
                      const float* __restrict__ Aw,
                      float* __restrict__ out) {
    extern __shared__ float sx[];          // 2 * PLANEF floats

    const int y  = blockIdx.x;
    const int z  = blockIdx.y;
    const int b  = blockIdx.z >> 1;
    const int xb = (blockIdx.z & 1) * XT;  // x base of this block's tile
    const int tid  = threadIdx.x;
    const int lane = tid & 31;
    const int wave = tid >> 5;
    const int wn = wave & 1;               // N tile -> x sub-base
    const int wm = wave >> 1;              // M tile -> o group
    const int nn = lane & 15;
    const int hi = lane >> 4;              // lane-half: K rows +2 / M rows +8
    const int lx = wn * 16 + nn;           // local output x (0..31)

    v8f acc = {};
    const float2* __restrict__ Aw2 = (const float2*)Aw;

    // issue async staging of plane dzp into buffer at float offset `bufo`
    auto stage = [&](int dzp, int bufo) {
        const int gz = z + dzp - 2;
        for (int idx = tid; idx < CCH * 5 * XW; idx += 128) {
            int ch = idx / (5 * XW);
            int r  = idx - ch * (5 * XW);
            int yy = r / XW;
            int xx = r - yy * XW;
            int gy = y + yy - 2;
            int gx = xb + xx - 2;
            // interleaved dest: [cpair][yy][xx][ch&1]
            float* ldst = &sx[bufo + ((((ch >> 1) * 5 + yy) * XW + xx) << 1) + (ch & 1)];
            if ((unsigned)gz < 64u && (unsigned)gy < 64u && (unsigned)gx < 64u)
                async_copy_b32(&x[(((size_t)(b * CCH + ch) * 64 + gz) * 64 + gy) * 64 + gx],
                               ldst);
            else
                *ldst = 0.f;
        }
    };

    stage(0, 0);
    wait_async0();
    __syncthreads();

    for (int dz = 0; dz < 5; ++dz) {
        const int cur = (dz & 1) * PLANEF;
        if (dz < 4) stage(dz + 1, ((dz + 1) & 1) * PLANEF);   // issue only

        for (int dy = 0; dy < 5; ++dy) {
            #pragma unroll
            for (int dx = 0; dx < 5; ++dx) {
                const int t = dz * 25 + dy * 5 + dx;
                // per-lane B base: cpair = 2c + hi, row dy, col lx + dx
                const float* __restrict__ bp =
                    &sx[cur + (((hi * 5 + dy) * XW + lx + dx) << 1)];
                const float2* __restrict__ ap = &Aw2[(t * 16 + wm) * 32 + lane];
                #pragma unroll
                for (int c = 0; c < 8; ++c) {
                    float2 af = ap[c * 64];                     // A(m,kk), A(m,kk+1)
                    float2 bf = *(const float2*)&bp[c * (2 * 5 * XW * 2)];
                    v2f a;  a.x = af.x;  a.y = af.y;
                    v2f bb; bb.x = bf.x; bb.y = bf.y;
                    acc = __builtin_amdgcn_wmma_f32_16x16x4_f32(
                        false, a, false, bb, (short)0, acc, false, false);
                }
            }
        }
        wait_async0();          // next plane resident before buffer swap
        __syncthreads();
    }

    // D layout: vgpr e -> (M=e, N=lane) lanes<16 ; (M=e+8, N=lane-16) lanes>=16
    const int obase = wm * 16 + hi * 8;
    const size_t base =
        ((((size_t)b * CCH + obase) * 64 + z) * 64 + y) * 64 + (xb + lx);
    #pragma unroll
    for (int e = 0; e < 8; ++e)
        out[base + (size_t)e * 64 * 64 * 64] = 0.1f * acc[e];
}

extern "C" void kernel_launch(void* const* d_in, const int* in_sizes, int n_in,
                              void* d_out, int out_size, void* d_ws, size_t ws_size,
                              hipStream_t stream) {
    const float* x    = (const float*)d_in[0];
    const float* tp   = (const float*)d_in[1];
    const float* wsc0 = (const float*)d_in[2];
    const float* wsc1 = (const float*)d_in[3];
    float* out = (float*)d_out;
    float* Aw  = (float*)d_ws;   // 125*8*2*32*2 floats = 512000 B

    build_weights_kernel<<<125, 256, 0, stream>>>(tp, wsc0, wsc1, Aw);
    conv_wmma_kernel<<<dim3(64, 64, 8), 128, 2 * PLANEF * sizeof(float), stream>>>(x, Aw, out);
}